// AttnDecoderRNN_3831110828073
// MI455X (gfx1250) — compile-verified
//
#include <hip/hip_runtime.h>
#include <math.h>

typedef float v2f __attribute__((ext_vector_type(2)));
typedef float v8f __attribute__((ext_vector_type(8)));

#define H_ 1024
#define V_ 50000
#define T_ 7
#define NTILE 3125                 /* V_/16 exactly */
#define K6_BLOCKS ((NTILE + 3) / 4)

/* ---------------- helpers ---------------- */
__device__ __forceinline__ float warp_sum(float v) {
  for (int off = 16; off; off >>= 1) v += __shfl_xor(v, off, 32);
  return v;
}
__device__ __forceinline__ float warp_max(float v) {
  for (int off = 16; off; off >>= 1) v = fmaxf(v, __shfl_xor(v, off, 32));
  return v;
}
__device__ __forceinline__ float sigf(float x) { return 1.f / (1.f + expf(-x)); }

__device__ __forceinline__ float dot4n(const float* __restrict__ w,
                                       const float* __restrict__ x,
                                       int n, int lane) {
  float acc = 0.f;
  for (int k = lane * 4; k < n; k += 128) {
    float4 a = *(const float4*)(w + k);
    float4 b = *(const float4*)(x + k);
    acc = fmaf(a.x, b.x, fmaf(a.y, b.y, fmaf(a.z, b.z, fmaf(a.w, b.w, acc))));
  }
  return warp_sum(acc);
}

/* ---------------- K0: embedding concat -> xcomb[0..1043] ---------------- */
__global__ void k0_embcomb(const int* __restrict__ input_id, const int* __restrict__ attr,
                           const float* __restrict__ emb, const float* __restrict__ wk,
                           const float* __restrict__ dd, float* __restrict__ xcomb) {
  int i = threadIdx.x;
  const float* erow = emb + (size_t)input_id[0] * H_;
  for (int k = i; k < H_; k += blockDim.x) xcomb[k] = erow[k];
  if (i < 10)       xcomb[H_ + i] = wk[attr[0] * 10 + i];
  else if (i < 20)  xcomb[H_ + i] = dd[attr[1] * 10 + (i - 10)];
}

/* ---------- K1: wd = WD*[h;c]+b, ench = UD*enc+b, whhp = Whh*h0 ---------- */
__global__ __launch_bounds__(256) void k1_gemvs(
    const float* __restrict__ WD_w, const float* __restrict__ WD_b,
    const float* __restrict__ hid,  const float* __restrict__ cel,
    const float* __restrict__ UD_w, const float* __restrict__ UD_b,
    const float* __restrict__ enc,  const float* __restrict__ Whh,
    float* __restrict__ wd, float* __restrict__ ench, float* __restrict__ whhp) {
  const int lane = threadIdx.x & 31;
  const int g = blockIdx.x * 8 + (threadIdx.x >> 5);
  if (g < 1024) {                       /* wd: 2048-dot against [h0|c0] */
    const float* w = WD_w + (size_t)g * 2048;
    float acc = 0.f;
    for (int k = lane * 4; k < 2048; k += 128) {
      float4 a = *(const float4*)(w + k);
      float4 b = (k < 1024) ? *(const float4*)(hid + k)
                            : *(const float4*)(cel + (k - 1024));
      acc = fmaf(a.x, b.x, fmaf(a.y, b.y, fmaf(a.z, b.z, fmaf(a.w, b.w, acc))));
    }
    acc = warp_sum(acc);
    if (lane == 0) wd[g] = acc + WD_b[g];
  } else if (g < 1024 + T_ * 1024) {    /* ench[t][i] */
    int q = g - 1024, t = q >> 10, i = q & 1023;
    float acc = dot4n(UD_w + (size_t)i * H_, enc + (size_t)t * H_, H_, lane);
    if (lane == 0) ench[q] = acc + UD_b[i];
  } else if (g < 12288) {               /* whhp[j] = Whh[j,:]·h0 */
    int j = g - 8192;
    float acc = dot4n(Whh + (size_t)j * H_, hid, H_, lane);
    if (lane == 0) whhp[j] = acc;
  }
}

/* ---------- K2: scores -> softmax -> ctx -> xcomb[1044..2067] ---------- */
__global__ __launch_bounds__(256) void k2_attn(
    const float* __restrict__ wd, const float* __restrict__ ench,
    const float* __restrict__ VD_w, const float* __restrict__ VD_b,
    const float* __restrict__ enc, float* __restrict__ xcomb,
    float* __restrict__ d_out) {
  __shared__ float red[256];
  __shared__ float aw[T_];
  const int tid = threadIdx.x;
  for (int t = 0; t < T_; t++) {
    float s = 0.f;
    for (int i = tid; i < H_; i += 256) s += VD_w[i] * tanhf(wd[i] + ench[t * H_ + i]);
    red[tid] = s; __syncthreads();
    for (int h = 128; h; h >>= 1) { if (tid < h) red[tid] += red[tid + h]; __syncthreads(); }
    if (tid == 0) aw[t] = red[0] + VD_b[0];
    __syncthreads();
  }
  if (tid == 0) {
    float m = -3.4e38f;
    for (int t = 0; t < T_; t++) m = fmaxf(m, aw[t]);
    float e[T_], s = 0.f;
    for (int t = 0; t < T_; t++) { e[t] = expf(aw[t] - m); s += e[t]; }
    for (int t = 0; t < T_; t++) { aw[t] = e[t] / s; d_out[V_ + 2 * H_ + t] = aw[t]; }
  }
  __syncthreads();
  for (int i = tid; i < H_; i += 256) {
    float c = 0.f;
    for (int t = 0; t < T_; t++) c = fmaf(aw[t], enc[t * H_ + i], c);
    xcomb[H_ + 20 + i] = c;
  }
}

/* ---------------- K3: x = relu(AC_w · xcomb + AC_b) ---------------- */
__global__ __launch_bounds__(256) void k3_ac(const float* __restrict__ AC_w,
                                             const float* __restrict__ AC_b,
                                             const float* __restrict__ xcomb,
                                             float* __restrict__ xv) {
  const int lane = threadIdx.x & 31;
  const int i = blockIdx.x * 8 + (threadIdx.x >> 5);
  if (i >= H_) return;
  float acc = dot4n(AC_w + (size_t)i * 2068, xcomb, 2068, lane);
  if (lane == 0) xv[i] = fmaxf(acc + AC_b[i], 0.f);
}

/* ---------------- K4: gates = Wih·x + bih + bhh + Whh·h0 ---------------- */
__global__ __launch_bounds__(256) void k4_gates(const float* __restrict__ Wih,
                                                const float* __restrict__ bih,
                                                const float* __restrict__ bhh,
                                                const float* __restrict__ whhp,
                                                const float* __restrict__ xv,
                                                float* __restrict__ gates) {
  const int lane = threadIdx.x & 31;
  const int j = blockIdx.x * 8 + (threadIdx.x >> 5);
  if (j >= 4 * H_) return;
  float acc = dot4n(Wih + (size_t)j * H_, xv, H_, lane);
  if (lane == 0) gates[j] = acc + bih[j] + bhh[j] + whhp[j];
}

/* ---------------- K5: LSTM pointwise -> h1, c1 ---------------- */
__global__ void k5_lstm(const float* __restrict__ gates, const float* __restrict__ cell0,
                        float* __restrict__ d_out, float* __restrict__ hvec) {
  const int i = threadIdx.x;  /* 1024 threads */
  float ig = sigf(gates[i]);
  float fg = sigf(gates[H_ + i]);
  float gg = tanhf(gates[2 * H_ + i]);
  float og = sigf(gates[3 * H_ + i]);
  float c1 = fg * cell0[i] + ig * gg;
  float h1 = og * tanhf(c1);
  d_out[V_ + i] = h1;
  d_out[V_ + H_ + i] = c1;
  hvec[i] = h1;
}

/* -------- K6: logits via V_WMMA_F32_16X16X4_F32, 16 vocab rows / wave --------
 * A (16x4 f32): lane m (0-15) holds row m K={kk,kk+1}; lane m+16 holds K={kk+2,kk+3}
 *   -> one global_load_b64 per lane; rows get contiguous 16B per instruction.
 * B (4x16 f32): B[k][n] = h[kk+k] for all n -> per-lane ds_load_b64 at
 *   sh_h + kk + 2*half gives exactly (v0,v1); no selects needed.
 * D: all N columns equal; lane 0 holds rows 0..7, lane 16 rows 8..15.
 */
#define WMMA_STEP(CREG, KK)                                                        \
  do {                                                                             \
    float2 a2 = *(const float2*)(wrow + (KK));                                     \
    float2 b2 = *(const float2*)(sh_b + (KK));                                     \
    v2f A; A.x = a2.x; A.y = a2.y;                                                 \
    v2f B; B.x = b2.x; B.y = b2.y;                                                 \
    CREG = __builtin_amdgcn_wmma_f32_16x16x4_f32(false, A, false, B, (short)0,     \
                                                 CREG, false, false);              \
  } while (0)

__global__ __launch_bounds__(128) void k6_logits(const float* __restrict__ out_w,
                                                 const float* __restrict__ out_b,
                                                 const float* __restrict__ hvec,
                                                 float* __restrict__ logits,
                                                 float* __restrict__ blockmax) {
  __shared__ float sh_h[H_];
  __shared__ float sh_m[4];
  const int tid = threadIdx.x;
  for (int i = tid; i < H_; i += 128) sh_h[i] = hvec[i];
  __syncthreads();

  const int wave = tid >> 5, lane = tid & 31;
  const int tile = blockIdx.x * 4 + wave;
  float mloc = -3.4e38f;
  if (tile < NTILE) {                    /* wave-uniform: EXEC all-ones for WMMA */
    const int r = lane & 15, half = lane >> 4;
    const int v0 = tile * 16;
    const float* wrow = out_w + (size_t)(v0 + r) * H_ + (half << 1);
    const float* sh_b = sh_h + (half << 1);       /* per-lane B base in LDS */
    v8f c0 = {}; v8f c1 = {}; v8f c2 = {}; v8f c3 = {};
    for (int k0 = 0; k0 < H_; k0 += 16) {
      WMMA_STEP(c0, k0);
      WMMA_STEP(c1, k0 + 4);
      WMMA_STEP(c2, k0 + 8);
      WMMA_STEP(c3, k0 + 12);
    }
    v8f cc = (c0 + c1) + (c2 + c3);
    /* lane 0 holds D rows 0..7 (any column), lane 16 holds rows 8..15 */
    if (r == 0) {
      const int base = v0 + (half << 3);
      float m = -3.4e38f;
#pragma unroll
      for (int j = 0; j < 8; j++) {
        float val = cc[j] + out_b[base + j];
        logits[base + j] = val;
        m = fmaxf(m, val);
      }
      mloc = m;
    }
  }
  mloc = warp_max(mloc);
  if (lane == 0) sh_m[wave] = mloc;
  __syncthreads();
  if (tid == 0)
    blockmax[blockIdx.x] = fmaxf(fmaxf(sh_m[0], sh_m[1]), fmaxf(sh_m[2], sh_m[3]));
}

/* ---------------- K7: global max of block maxima ---------------- */
__global__ void k7_maxred(const float* __restrict__ bm, int n, float* __restrict__ scal) {
  __shared__ float red[256];
  const int tid = threadIdx.x;
  float m = -3.4e38f;
  for (int i = tid; i < n; i += 256) m = fmaxf(m, bm[i]);
  red[tid] = m; __syncthreads();
  for (int h = 128; h; h >>= 1) { if (tid < h) red[tid] = fmaxf(red[tid], red[tid + h]); __syncthreads(); }
  if (tid == 0) scal[0] = red[0];
}

/* ---------------- K8: per-block partial sum of exp(x - M) ---------------- */
__global__ __launch_bounds__(256) void k8_psum(const float* __restrict__ logits,
                                               const float* __restrict__ scal,
                                               float* __restrict__ psum) {
  __shared__ float red[256];
  const int tid = threadIdx.x;
  const float M = scal[0];
  float s = 0.f;
  for (int i = blockIdx.x * 256 + tid; i < V_; i += 256 * 256) s += expf(logits[i] - M);
  red[tid] = s; __syncthreads();
  for (int h = 128; h; h >>= 1) { if (tid < h) red[tid] += red[tid + h]; __syncthreads(); }
  if (tid == 0) psum[blockIdx.x] = red[0];
}

/* ---------------- K9: logZ = M + log(sum) ---------------- */
__global__ void k9_logz(const float* __restrict__ psum, float* __restrict__ scal) {
  __shared__ float red[256];
  const int tid = threadIdx.x;
  red[tid] = psum[tid]; __syncthreads();
  for (int h = 128; h; h >>= 1) { if (tid < h) red[tid] += red[tid + h]; __syncthreads(); }
  if (tid == 0) scal[1] = scal[0] + logf(red[0]);
}

/* ---------------- K10: output = logits - logZ ---------------- */
__global__ void k10_out(const float* __restrict__ logits, const float* __restrict__ scal,
                        float* __restrict__ out) {
  const int i = blockIdx.x * 256 + threadIdx.x;
  if (i < V_) out[i] = logits[i] - scal[1];
}

/* ---------------- launcher ---------------- */
extern "C" void kernel_launch(void* const* d_in, const int* in_sizes, int n_in,
                              void* d_out_v, int out_size, void* d_ws, size_t ws_size,
                              hipStream_t stream) {
  (void)in_sizes; (void)n_in; (void)out_size; (void)ws_size;
  const int*   input_id = (const int*)d_in[0];
  const int*   attr     = (const int*)d_in[1];
  const float* hidden   = (const float*)d_in[2];
  const float* cell     = (const float*)d_in[3];
  const float* enc      = (const float*)d_in[4];
  const float* emb      = (const float*)d_in[5];
  const float* wk_emb   = (const float*)d_in[6];
  const float* dd_emb   = (const float*)d_in[7];
  const float* WD_w     = (const float*)d_in[8];
  const float* WD_b     = (const float*)d_in[9];
  const float* UD_w     = (const float*)d_in[10];
  const float* UD_b     = (const float*)d_in[11];
  const float* VD_w     = (const float*)d_in[12];
  const float* VD_b     = (const float*)d_in[13];
  const float* AC_w     = (const float*)d_in[14];
  const float* AC_b     = (const float*)d_in[15];
  const float* Wih      = (const float*)d_in[16];
  const float* Whh      = (const float*)d_in[17];
  const float* bih      = (const float*)d_in[18];
  const float* bhh      = (const float*)d_in[19];
  const float* out_w    = (const float*)d_in[20];
  const float* out_b    = (const float*)d_in[21];
  float* out = (float*)d_out_v;
  float* ws  = (float*)d_ws;

  /* workspace layout (floats, all 16B-aligned offsets) */
  float* logits   = ws + 0;       /* 50000 */
  float* blockmax = ws + 50000;   /* 800   */
  float* psum     = ws + 50800;   /* 256   */
  float* scal     = ws + 51056;   /* 8: [0]=M, [1]=logZ */
  float* xcomb    = ws + 51064;   /* 2068  */
  float* wd       = ws + 53136;   /* 1024  */
  float* ench     = ws + 54160;   /* 7168  */
  float* whhp     = ws + 61328;   /* 4096  */
  float* xv       = ws + 65424;   /* 1024  */
  float* gates    = ws + 66448;   /* 4096  */
  float* hvec     = ws + 70544;   /* 1024  */

  k0_embcomb<<<1, 256, 0, stream>>>(input_id, attr, emb, wk_emb, dd_emb, xcomb);
  k1_gemvs<<<1536, 256, 0, stream>>>(WD_w, WD_b, hidden, cell, UD_w, UD_b, enc, Whh,
                                     wd, ench, whhp);
  k2_attn<<<1, 256, 0, stream>>>(wd, ench, VD_w, VD_b, enc, xcomb, out);
  k3_ac<<<128, 256, 0, stream>>>(AC_w, AC_b, xcomb, xv);
  k4_gates<<<512, 256, 0, stream>>>(Wih, bih, bhh, whhp, xv, gates);
  k5_lstm<<<1, 1024, 0, stream>>>(gates, cell, out, hvec);
  k6_logits<<<K6_BLOCKS, 128, 0, stream>>>(out_w, out_b, hvec, logits, blockmax);
  k7_maxred<<<1, 256, 0, stream>>>(blockmax, K6_BLOCKS, scal);
  k8_psum<<<256, 256, 0, stream>>>(logits, scal, psum);
  k9_logz<<<1, 256, 0, stream>>>(psum, scal);
  k10_out<<<(V_ + 255) / 256, 256, 0, stream>>>(logits, scal, out);
}